// CustomMinkowskiChannelwiseConvolution_8177617732131
// MI455X (gfx1250) — compile-verified
//
#include <hip/hip_runtime.h>
#include <stdint.h>

#define C 32
#define KVOL 27
#define EPW 32          // edges per wave (contiguous chunk)
#define BLOCK 256       // 8 waves

__global__ __launch_bounds__(256) void mink_zero_f4(float4* __restrict__ out, int n4) {
    int i = blockIdx.x * blockDim.x + threadIdx.x;
    if (i < n4) out[i] = make_float4(0.f, 0.f, 0.f, 0.f);
}

__global__ __launch_bounds__(256) void mink_zero_tail(float* __restrict__ out, int base, int n) {
    int i = base + blockIdx.x * blockDim.x + threadIdx.x;
    if (i < n) out[i] = 0.f;
}

// One wave (32 lanes) per edge; lane == channel. Each wave owns a contiguous
// chunk of EPW edges (edges are sorted by (kernel offset, out_idx), so the
// chunk's atomics sweep ascending addresses -> minimal contention).
// Wave-uniform index/coord fetches go through the scalar path (readfirstlane
// -> s_load); weights staged to LDS via CDNA5 async-to-LDS DMA; scatter via
// native no-return f32 global atomic.
__global__ __launch_bounds__(256) void mink_dwconv_scatter(
    const int*   __restrict__ coords,    // [N,3]
    const int*   __restrict__ in_idx,    // [E]
    const int*   __restrict__ out_idx,   // [E]
    const float* __restrict__ in_feats,  // [N,C]
    const float* __restrict__ kern,      // [27,C]
    float*       __restrict__ out,       // [N,C]
    int E)
{
    __shared__ __align__(16) float s_w[KVOL * C];   // 3456 B

    const int t = threadIdx.x;

    // Stage 27x32 f32 weights into LDS: 216 lanes x 16 B async b128 copies.
    if (t < (KVOL * C * 4) / 16) {
        uint32_t lds_addr = (uint32_t)(uintptr_t)(&s_w[0]) + (uint32_t)(t * 16);
        uint32_t goff     = (uint32_t)(t * 16);
        asm volatile("global_load_async_to_lds_b128 %0, %1, %2"
                     :
                     : "v"(lds_addr), "v"(goff), "s"(kern)
                     : "memory");
    }
    asm volatile("s_wait_asynccnt 0" ::: "memory");
    __syncthreads();

    const int lane = t & 31;
    const int wave = blockIdx.x * (BLOCK >> 5) + (t >> 5);

    // Force the chunk base into an SGPR so index/coord loads become scalar.
    const int e0 = __builtin_amdgcn_readfirstlane(wave * EPW);
    int e1 = e0 + EPW;
    if (e1 > E) e1 = E;

#pragma unroll 4
    for (int e = e0; e < e1; ++e) {
        const int ii = in_idx[e];    // uniform -> s_load
        const int oo = out_idx[e];   // uniform -> s_load

        // k1d from coordinate delta: 9*(dx+1) + 3*(dy+1) + (dz+1)   (SALU)
        const int cix = coords[3 * ii + 0];
        const int ciy = coords[3 * ii + 1];
        const int ciz = coords[3 * ii + 2];
        const int cox = coords[3 * oo + 0];
        const int coy = coords[3 * oo + 1];
        const int coz = coords[3 * oo + 2];
        const int k   = 9 * (cix - cox + 1) + 3 * (ciy - coy + 1) + (ciz - coz + 1);

        const float v = in_feats[(size_t)ii * C + lane];  // 128 B coalesced gather
        const float w = s_w[k * C + lane];                // conflict-free LDS read
        float p = v * w;

        float* addr = out + (size_t)oo * C + lane;        // 128 B coalesced scatter
        asm volatile("global_atomic_add_f32 %0, %1, off"
                     :
                     : "v"(addr), "v"(p)
                     : "memory");
    }
}

extern "C" void kernel_launch(void* const* d_in, const int* in_sizes, int n_in,
                              void* d_out, int out_size, void* d_ws, size_t ws_size,
                              hipStream_t stream) {
    const int*   coords   = (const int*)d_in[0];
    const int*   in_idx   = (const int*)d_in[1];
    const int*   out_idx  = (const int*)d_in[2];
    const float* in_feats = (const float*)d_in[3];
    const float* kern     = (const float*)d_in[4];
    float*       out      = (float*)d_out;

    const int E = in_sizes[1];

    // Zero the output (accumulated via atomics afterwards).
    const int n4  = out_size / 4;
    const int rem = out_size - n4 * 4;
    if (n4 > 0) {
        mink_zero_f4<<<(n4 + BLOCK - 1) / BLOCK, BLOCK, 0, stream>>>((float4*)out, n4);
    }
    if (rem > 0) {
        mink_zero_tail<<<1, BLOCK, 0, stream>>>(out, n4 * 4, out_size);
    }

    if (E > 0) {
        const int edges_per_block = (BLOCK / 32) * EPW;  // 256 edges per block
        const int blocks = (E + edges_per_block - 1) / edges_per_block;
        mink_dwconv_scatter<<<blocks, BLOCK, 0, stream>>>(
            coords, in_idx, out_idx, in_feats, kern, out, E);
    }
}